// calc_divergence_4406636445920
// MI455X (gfx1250) — compile-verified
//
#include <hip/hip_runtime.h>

// Divergence of a periodic (2,3,256,256,256) fp32 velocity field.
// Memory-bound: ~537 MB min HBM traffic -> ~23 us floor at 23.3 TB/s.
// Strategy:
//  - u rows DMA'd global->LDS via GLOBAL_LOAD_ASYNC_TO_LDS_B128 (ASYNCcnt),
//    x-halo reads then come from LDS (u read from HBM exactly once).
//  - v/w: direct global_load_b128, overlapped with the async DMA.
//  - periodic wrap via bitmask (power-of-2 dims), no div/mod.
//  - non-temporal global_store_b128 for write-once output (preserve L2 for
//    the x2-reused v/w rows), global_prefetch_b8 one block ahead.

typedef float v4f __attribute__((ext_vector_type(4)));

#define DIM   256u
#define DMASK 255u

__global__ __launch_bounds__(256, 1)
void div_kernel(const float* __restrict__ x, float* __restrict__ out) {
    // 4 complete u-rows per block (block = 256 threads = 4 rows x 64 float4)
    __shared__ __align__(16) float lds_u[4 * DIM];

    const unsigned tid = threadIdx.x;
    // Global float4 index: bits [5:0]=x-vec, [13:6]=y, [21:14]=z, [22]=batch
    const unsigned g  = blockIdx.x * 256u + tid;
    const unsigned xv = g & 63u;
    const unsigned y  = (g >> 6) & DMASK;   // block's 4 rows never wrap (4 | 256)
    const unsigned z  = (g >> 14) & DMASK;
    const unsigned b  = g >> 22;
    const unsigned x0 = xv << 2;
    const unsigned r  = tid >> 6;           // local row 0..3

    const size_t vol = (size_t)DIM * DIM * DIM;
    const float* __restrict__ u = x + (size_t)b * 3u * vol;
    const float* __restrict__ v = u + vol;
    const float* __restrict__ w = v + vol;

    const size_t zrow  = (size_t)z * DIM;
    const size_t rowzy = (zrow + y) * DIM;

    // ---- issue async DMA of this block's u rows: global -> LDS (gfx1250) ----
    {
        const unsigned long long gsrc =
            (unsigned long long)(const void*)(u + rowzy + x0);
        // Generic LDS pointer: low 32 bits == workgroup-relative LDS byte addr.
        const unsigned lds_byte =
            (unsigned)(unsigned long long)(const void*)&lds_u[r * DIM + x0];
        asm volatile("global_load_async_to_lds_b128 %0, %1, off"
                     :
                     : "v"(lds_byte), "v"(gsrc)
                     : "memory");
    }

    // Warm L2 one block ahead (next block DMAs rows y+4..y+7 of same z):
    __builtin_prefetch(u + (zrow + ((y + 4u) & DMASK)) * DIM + x0, 0, 0);

    // ---- dv / dw global loads overlap the async DMA ----
    const unsigned yp = (y + 1u)    & DMASK;
    const unsigned ym = (y + DMASK) & DMASK;
    const unsigned zp = (z + 1u)    & DMASK;
    const unsigned zm = (z + DMASK) & DMASK;

    const v4f vp = *(const v4f*)(v + (zrow + yp) * DIM + x0);
    const v4f vm = *(const v4f*)(v + (zrow + ym) * DIM + x0);
    const v4f wp = *(const v4f*)(w + ((size_t)zp * DIM + yp) * DIM + x0);
    const v4f wm = *(const v4f*)(w + ((size_t)zm * DIM + ym) * DIM + x0);

    // ---- wait for the DMA, make all 4 rows visible block-wide ----
    asm volatile("s_wait_asynccnt 0" ::: "memory");
    __syncthreads();

    // ---- du: central difference along x, halo from LDS (periodic) ----
    const float* lrow = &lds_u[r * DIM];
    const v4f  uc = *(const v4f*)(lrow + x0);         // ds_load_b128
    const float ul = lrow[(x0 + DMASK) & DMASK];      // ds_load_b32
    const float ur = lrow[(x0 + 4u)   & DMASK];       // ds_load_b32

    v4f rr;
    rr.x = 0.5f * ((uc.y - ul)   + (vp.x - vm.x) + (wp.x - wm.x));
    rr.y = 0.5f * ((uc.z - uc.x) + (vp.y - vm.y) + (wp.y - wm.y));
    rr.z = 0.5f * ((uc.w - uc.y) + (vp.z - vm.z) + (wp.z - wm.z));
    rr.w = 0.5f * ((ur   - uc.z) + (vp.w - vm.w) + (wp.w - wm.w));

    // Write-once output: non-temporal store keeps L2 for reused v/w rows.
    v4f* o = (v4f*)(out + (size_t)b * vol + rowzy + x0);
    __builtin_nontemporal_store(rr, o);
}

extern "C" void kernel_launch(void* const* d_in, const int* in_sizes, int n_in,
                              void* d_out, int out_size, void* d_ws, size_t ws_size,
                              hipStream_t stream) {
    (void)in_sizes; (void)n_in; (void)out_size; (void)d_ws; (void)ws_size;
    const float* x = (const float*)d_in[0];
    float* out = (float*)d_out;

    // 2 batches * 256^3 elements / 4 per thread = 8,388,608 threads
    const unsigned nvec   = 2u * DIM * DIM * (DIM / 4u);
    const unsigned blocks = nvec / 256u;   // 32768 blocks, 8 wave32 each
    div_kernel<<<dim3(blocks), dim3(256), 0, stream>>>(x, out);
}